// DiagonalMatrix_3917010174744
// MI455X (gfx1250) — compile-verified
//
#include <hip/hip_runtime.h>
#include <cstdint>

// ---------------------------------------------------------------------------
// DiagonalMatrix: out[r, j] = x[r, j] * w[j]   (ROWS=8192, N=4096, f32)
//
// Memory-bound: 268 MB moved vs 33.5 MFLOP -> AI = 0.125 FLOP/B.
// Floor at 23.3 TB/s HBM is ~11.5 us. WMMA deliberately NOT used (x16 wasted
// FLOPs + layout traffic for a streaming op). Instead we exercise the CDNA5
// data-movement paths: b128 vector accesses, NT temporal hints on the big
// stream, and async global->LDS staging of the weight row (ASYNCcnt path).
// ---------------------------------------------------------------------------

typedef float float4v __attribute__((ext_vector_type(4)));
typedef int   int4v   __attribute__((ext_vector_type(4)));

#define AS_GLOBAL __attribute__((address_space(1)))
#define AS_LOCAL  __attribute__((address_space(3)))

__global__ __launch_bounds__(256) void diag_scale_kernel(
    const float* __restrict__ x,
    const float* __restrict__ w,
    float* __restrict__ out,
    int rows)
{
    constexpr int N        = 4096;        // columns (fixed by reference)
    constexpr int NV       = N / 4;       // 1024 float4 per row
    constexpr int TPB      = 256;         // 8 wave32s per block
    constexpr int V_PER_T  = NV / TPB;    // 4 float4 per thread per row
    constexpr int ROWS_PER_BLOCK = 8;     // amortize weight staging

    __shared__ float lds_w[N];            // 16 KB of the 320 KB WGP LDS

    const int t = threadIdx.x;

    // ---- Stage weight row into LDS via CDNA5 async global->LDS copies ----
#if defined(__gfx1250__) && __has_builtin(__builtin_amdgcn_global_load_async_to_lds_b128)
    {
        AS_GLOBAL char* gsrc = (AS_GLOBAL char*)w;   // const cast: builtin takes non-const ptr
        AS_LOCAL  char* ldst = (AS_LOCAL  char*)lds_w;
        #pragma unroll
        for (int k = 0; k < V_PER_T; ++k) {
            const int byteOff = (t + k * TPB) * 16;
            __builtin_amdgcn_global_load_async_to_lds_b128(
                (AS_GLOBAL int4v*)(gsrc + byteOff),
                (AS_LOCAL  int4v*)(ldst + byteOff),
                /*offset=*/0, /*cpol=*/0);
        }
#if __has_builtin(__builtin_amdgcn_s_wait_asynccnt)
        __builtin_amdgcn_s_wait_asynccnt(0);
#else
        asm volatile("s_wait_asynccnt 0" ::: "memory");
#endif
    }
#else
    // Fallback: plain b128 copy through VGPRs.
    #pragma unroll
    for (int k = 0; k < V_PER_T; ++k) {
        const int v = t + k * TPB;
        ((float4v*)lds_w)[v] = ((const float4v*)w)[v];
    }
#endif
    __syncthreads();

    // Each thread's 4 weight float4s live in registers for all 8 rows.
    float4v wr[V_PER_T];
    #pragma unroll
    for (int k = 0; k < V_PER_T; ++k)
        wr[k] = ((const float4v*)lds_w)[t + k * TPB];

    const int row0 = blockIdx.x * ROWS_PER_BLOCK;
    #pragma unroll
    for (int r = 0; r < ROWS_PER_BLOCK; ++r) {
        const int row = row0 + r;
        if (row >= rows) break;
        const float4v* __restrict__ xrow = (const float4v*)(x   + (size_t)row * N);
        float4v*       __restrict__ orow = (float4v*)      (out + (size_t)row * N);
        #pragma unroll
        for (int k = 0; k < V_PER_T; ++k) {
            const int v = t + k * TPB;
            // Streamed exactly once: non-temporal load + non-temporal store
            // so the 256 MB stream does not evict the resident weight row.
            float4v xv = __builtin_nontemporal_load(&xrow[v]);
            float4v ov = xv * wr[k];
            __builtin_nontemporal_store(ov, &orow[v]);
        }
    }
}

extern "C" void kernel_launch(void* const* d_in, const int* in_sizes, int n_in,
                              void* d_out, int out_size, void* d_ws, size_t ws_size,
                              hipStream_t stream) {
    const float* x = (const float*)d_in[0];   // (ROWS, N) f32
    const float* w = (const float*)d_in[1];   // (N,) f32
    float* out     = (float*)d_out;           // (ROWS, N) f32

    constexpr int N = 4096;
    constexpr int ROWS_PER_BLOCK = 8;
    const int rows = in_sizes[0] / N;         // 8192
    const int grid = (rows + ROWS_PER_BLOCK - 1) / ROWS_PER_BLOCK;  // 1024

    diag_scale_kernel<<<grid, 256, 0, stream>>>(x, w, out, rows);
}